// ENMF_58299886076627
// MI455X (gfx1250) — compile-verified
//
#include <hip/hip_runtime.h>

// CDNA5 / gfx1250. wave32. ENMF loss:
//   out = 0.1 * sum(part1 * part2 * h h^T) + sum(0.9*hpq^2 - 2*hpq)
// part1/part2 are 64x64 Gram matrices -> V_WMMA_F32_16X16X4_F32 (full f32 path).
// hpq is an L2-resident gather (item table 25.6MB << 192MB L2) + 64-dot per row.

typedef __attribute__((ext_vector_type(2))) float v2f;
typedef __attribute__((ext_vector_type(8))) float v8f;

#define ITEM_NUM   100000
#define DIMS       64
#define BATCH      4096
#define LLEN       200

// ws layout (floats):
//   [0,4096)        part1 accumulator (64x64)
//   [4096,8192)     part2 accumulator (64x64)
//   [8192]          pos_data_loss accumulator
//   [8448, +262144) gathered u_emb [4096 x 64]
#define WS_PART1  0
#define WS_PART2  4096
#define WS_LOSS   8192
#define WS_UEMB   8448

__global__ void init_ws_kernel(float* ws) {
    int i = blockIdx.x * blockDim.x + threadIdx.x;
    if (i < WS_UEMB) ws[i] = 0.0f;
}

// Load the 4 column-block fragments for one K=4 chunk. off is a 32-bit element
// offset (uniform base W + per-lane voffset -> saddr-form global_load_b32 with
// immediate offsets, clause-grouped). Lane l<16 supplies K0/K1, l>=16 K2/K3.
__device__ __forceinline__ void load_frags(const float* __restrict__ W,
                                           unsigned off, v2f f[4]) {
#pragma unroll
    for (int i = 0; i < 4; ++i) {
        f[i][0] = W[off + i * 16];          // row r0,   col 16i+m
        f[i][1] = W[off + DIMS + i * 16];   // row r0+1, col 16i+m
    }
}

// 16 WMMAs: frag[i] serves as A for row-block i and B for col-block i (Gram).
__device__ __forceinline__ void mma_all(v8f acc[4][4], const v2f f[4]) {
#pragma unroll
    for (int ti = 0; ti < 4; ++ti)
#pragma unroll
        for (int tj = 0; tj < 4; ++tj)
            acc[ti][tj] = __builtin_amdgcn_wmma_f32_16x16x4_f32(
                false, f[ti], false, f[tj], (short)0, acc[ti][tj], false, false);
}

// C[64,64] += W[0:n,:]^T W[0:n,:] via 4x4 grid of 16x16 f32 WMMA tiles.
__global__ void __launch_bounds__(256) gram64_wmma_kernel(
        const float* __restrict__ W, int n_rows, float* __restrict__ out) {
    __shared__ float lds_acc[DIMS * DIMS];
    for (int i = threadIdx.x; i < DIMS * DIMS; i += blockDim.x) lds_acc[i] = 0.0f;
    __syncthreads();

    const int lane  = threadIdx.x & 31;
    const int wave  = threadIdx.x >> 5;
    const int wpwg  = blockDim.x >> 5;
    // lane-invariant by construction; force into SGPR so the pipeline loop
    // control is pure SALU (s_cmp/s_cbranch, no per-iteration EXEC churn).
    const int gwave = __builtin_amdgcn_readfirstlane(blockIdx.x * wpwg + wave);
    const int nwav  = gridDim.x * wpwg;

    const unsigned m    = lane & 15;                 // column within a 16-block
    const unsigned ro   = (unsigned)(lane >> 4) << 1;// K-row offset: 0 or 2
    const unsigned loff = ro * DIMS + m;             // per-lane element offset

    v8f acc[4][4];
#pragma unroll
    for (int i = 0; i < 4; ++i)
#pragma unroll
        for (int j = 0; j < 4; ++j)
#pragma unroll
            for (int k = 0; k < 8; ++k) acc[i][j][k] = 0.0f;

    // ---- main loop: full K=4 chunks, no predication, 2-stage pipeline ----
    const int full = n_rows >> 2;
    int c = gwave;
    if (c < full) {
        v2f cur[4];
        load_frags(W, (unsigned)c * (4 * DIMS) + loff, cur);
        for (;;) {
            const int  cn   = c + nwav;
            const bool more = cn < full;             // scalar -> s_cbranch
            v2f nxt[4];
            if (more) load_frags(W, (unsigned)cn * (4 * DIMS) + loff, nxt);
            mma_all(acc, cur);
            if (!more) break;
#pragma unroll
            for (int i = 0; i < 4; ++i) cur[i] = nxt[i];
            c = cn;
        }
    }

    // ---- tail: at most 3 rows, handled once by global wave 0 ----
    if ((n_rows & 3) && gwave == 0) {
        const int r0 = (full << 2) + (int)ro;
        v2f t[4];
#pragma unroll
        for (int i = 0; i < 4; ++i) {
            const int col = i * 16 + (int)m;
            t[i][0] = (r0     < n_rows) ? W[r0 * DIMS + col]       : 0.0f;
            t[i][1] = (r0 + 1 < n_rows) ? W[(r0 + 1) * DIMS + col] : 0.0f;
        }
        mma_all(acc, t);
    }

    // C/D layout: VGPR p -> row 16*ti + p (+8 for lanes 16..31), col 16*tj + (lane&15)
    const int crow = (lane >> 4) << 3;
    const int ccol = lane & 15;
#pragma unroll
    for (int ti = 0; ti < 4; ++ti)
#pragma unroll
        for (int tj = 0; tj < 4; ++tj)
#pragma unroll
            for (int p = 0; p < 8; ++p)
                atomicAdd(&lds_acc[(ti * 16 + crow + p) * DIMS + tj * 16 + ccol],
                          acc[ti][tj][p]);
    __syncthreads();
    for (int i = threadIdx.x; i < DIMS * DIMS; i += blockDim.x)
        atomicAdd(&out[i], lds_acc[i]);
}

// u_emb[b,:] = user_W[uids[b],:]; one float4 per thread (b128 loads/stores).
__global__ void gather_rows_kernel(const float* __restrict__ table,
                                   const int* __restrict__ idx,
                                   float* __restrict__ dst, int nrows) {
    int t = blockIdx.x * blockDim.x + threadIdx.x;
    if (t >= nrows * (DIMS / 4)) return;
    int b = t >> 4;                    // DIMS/4 == 16 float4 per row
    int c = t & 15;
    long row = idx[b];
    ((float4*)dst)[(long)b * 16 + c] = ((const float4*)table)[row * 16 + c];
}

// hpq[b,l] = sum_d u[b,d]*h[d]*item_W[iid,d] (masked at padding id)
// loss += 0.9*hpq^2 - 2*hpq. Block per b; wave per list entry; lane holds float2.
__global__ void __launch_bounds__(256) hpq_loss_kernel(
        const float* __restrict__ u_emb, const int* __restrict__ pos_iids,
        const float* __restrict__ item_W, const float* __restrict__ h,
        float* __restrict__ loss_acc) {
    __shared__ float wsum[8];
    const int b    = blockIdx.x;
    const int lane = threadIdx.x & 31;
    const int wave = __builtin_amdgcn_readfirstlane(threadIdx.x >> 5);

    const float uhx = u_emb[b * DIMS + lane * 2]     * h[lane * 2];
    const float uhy = u_emb[b * DIMS + lane * 2 + 1] * h[lane * 2 + 1];

    float acc = 0.0f;
    for (int l = wave; l < LLEN; l += 8) {          // scalar loop bounds
        // lane-invariant by construction; force scalar so the mask test is
        // s_cmp/s_cbranch and the row gather is saddr-based.
        const int iid = __builtin_amdgcn_readfirstlane(pos_iids[b * LLEN + l]);
        if (iid != ITEM_NUM) {
            const float2 rv =
                *(const float2*)(item_W + (unsigned)iid * DIMS + lane * 2);
            float s = rv.x * uhx + rv.y * uhy;
#pragma unroll
            for (int off = 16; off > 0; off >>= 1)
                s += __shfl_xor(s, off, 32);
            acc += 0.9f * s * s - 2.0f * s;        // identical in all lanes
        }
    }
    if (lane == 0) wsum[wave] = acc;
    __syncthreads();
    if (threadIdx.x == 0) {
        float t = 0.0f;
#pragma unroll
        for (int w = 0; w < 8; ++w) t += wsum[w];
        atomicAdd(loss_acc, t);
    }
}

__global__ void __launch_bounds__(256) finalize_kernel(
        const float* __restrict__ ws, const float* __restrict__ h,
        float* __restrict__ out) {
    __shared__ float red[256];
    const float* p1 = ws + WS_PART1;
    const float* p2 = ws + WS_PART2;
    float acc = 0.0f;
    for (int e = threadIdx.x; e < DIMS * DIMS; e += blockDim.x) {
        int i = e >> 6, j = e & 63;
        acc += p1[e] * p2[e] * h[i] * h[j];        // part3[i,j] = h[i]*h[j]
    }
    red[threadIdx.x] = acc;
    __syncthreads();
    for (int s = blockDim.x >> 1; s > 0; s >>= 1) {
        if (threadIdx.x < (unsigned)s) red[threadIdx.x] += red[threadIdx.x + s];
        __syncthreads();
    }
    if (threadIdx.x == 0) out[0] = 0.1f * red[0] + ws[WS_LOSS];
}

extern "C" void kernel_launch(void* const* d_in, const int* in_sizes, int n_in,
                              void* d_out, int out_size, void* d_ws, size_t ws_size,
                              hipStream_t stream) {
    const int*   uids     = (const int*)d_in[0];
    const int*   pos_iids = (const int*)d_in[1];
    const float* user_W   = (const float*)d_in[2];
    const float* item_W   = (const float*)d_in[3];
    const float* h        = (const float*)d_in[4];
    float* out = (float*)d_out;
    float* ws  = (float*)d_ws;

    float* part1 = ws + WS_PART1;
    float* part2 = ws + WS_PART2;
    float* loss  = ws + WS_LOSS;
    float* u_emb = ws + WS_UEMB;

    // 1) zero accumulators (fresh every call -> deterministic)
    init_ws_kernel<<<(WS_UEMB + 255) / 256, 256, 0, stream>>>(ws);
    // 2) part1 = item_W^T item_W over 100001 rows (WMMA f32, streams 25.6MB)
    gram64_wmma_kernel<<<128, 256, 0, stream>>>(item_W, ITEM_NUM + 1, part1);
    // 3) gather u_emb = user_W[uids]
    gather_rows_kernel<<<(BATCH * 16 + 255) / 256, 256, 0, stream>>>(
        user_W, uids, u_emb, BATCH);
    // 4) part2 = u_emb^T u_emb over 4096 rows (WMMA f32)
    gram64_wmma_kernel<<<32, 256, 0, stream>>>(u_emb, BATCH, part2);
    // 5) hpq loss: 819200 gathered rows, one row per wave-load
    hpq_loss_kernel<<<BATCH, 256, 0, stream>>>(u_emb, pos_iids, item_W, h, loss);
    // 6) scalar combine
    finalize_kernel<<<1, 256, 0, stream>>>(ws, h, out);
}